// rnnCell_spring__70824010711799
// MI455X (gfx1250) — compile-verified
//
#include <hip/hip_runtime.h>
#include <hip/hip_bf16.h>

#define SIM_DT (1.0f / 120.0f)

typedef __attribute__((ext_vector_type(2))) float v2f;
typedef __attribute__((ext_vector_type(4))) float v4f;
typedef __attribute__((ext_vector_type(8))) float v8f;

// Workspace layout (floats): Mbig[32*32] | carryA[32] | carryB[32]
// Mbig[2j+p][2i+q] = (A^(j-i) * W^T)[p][q]  (j>=i, else 0)
// carryA/B[2j+p]   = A^(j+1)[p][0/1]
__global__ void rnn_spring_setup_kernel(const float* __restrict__ k_w,
                                        const float* __restrict__ b_w,
                                        const float* __restrict__ input_mat,
                                        const float* __restrict__ input_mask,
                                        float* __restrict__ ws) {
    if (threadIdx.x != 0 || blockIdx.x != 0) return;
    const float k = fabsf(k_w[0]);
    const float b = fabsf(b_w[0]);
    float W[2][2];
    for (int i = 0; i < 2; ++i)
        for (int j = 0; j < 2; ++j)
            W[i][j] = input_mat[i * 2 + j] * input_mask[i * 2 + j] * SIM_DT;
    const float A[2][2] = {{1.0f, SIM_DT}, {-k * SIM_DT, 1.0f - b * SIM_DT}};
    float Ap[17][2][2];
    Ap[0][0][0] = 1.0f; Ap[0][0][1] = 0.0f; Ap[0][1][0] = 0.0f; Ap[0][1][1] = 1.0f;
    for (int n = 1; n <= 16; ++n)
        for (int p = 0; p < 2; ++p)
            for (int q = 0; q < 2; ++q)
                Ap[n][p][q] = A[p][0] * Ap[n - 1][0][q] + A[p][1] * Ap[n - 1][1][q];
    float* Mbig = ws;
    float* cA = ws + 1024;
    float* cB = ws + 1056;
    for (int j = 0; j < 16; ++j)
        for (int i = 0; i < 16; ++i)
            for (int p = 0; p < 2; ++p)
                for (int q = 0; q < 2; ++q) {
                    float v = 0.0f;
                    if (j >= i) {
                        const int d = j - i;
                        // (A^d * W^T)[p][q] = sum_m A^d[p][m] * W[q][m]
                        v = Ap[d][p][0] * W[q][0] + Ap[d][p][1] * W[q][1];
                    }
                    Mbig[(2 * j + p) * 32 + (2 * i + q)] = v;
                }
    for (int j = 0; j < 16; ++j)
        for (int p = 0; p < 2; ++p) {
            cA[2 * j + p] = Ap[j + 1][p][0];
            cB[2 * j + p] = Ap[j + 1][p][1];
        }
}

// One wave32 per 16-batch tile. Per 16-timestep chunk: forced response is a
// 32x32 (M) x 32x16 (u-tile) matmul via v_wmma_f32_16x16x4_f32 (two 16x16
// output tiles, 8 K-steps each), then a per-lane carry add of A^(j+1)*X_prev.
// Depth-2 software pipeline with a copy-free 3-chunk rotation; NT hints on
// both streams; global_prefetch ~5 chunks ahead.
__global__ __launch_bounds__(32) void rnn_spring_scan_kernel(
        const float* __restrict__ u, const float* __restrict__ x0,
        float* __restrict__ out, const float* __restrict__ ws, int T) {
    const int lane  = threadIdx.x & 31;
    const int laneM = lane & 15;
    const bool hi   = lane >= 16;
    const int tile  = blockIdx.x;               // one wave per block
    const int myBatch = tile * 16 + laneM;

    const float* __restrict__ Mbig = ws;
    const float* __restrict__ cA   = ws + 1024;
    const float* __restrict__ cB   = ws + 1056;

    // A-matrix fragments (constant across chunks): lane holds row = t*16+laneM,
    // VGPR pair = K {4ks+0,1} (lanes 0-15) or {4ks+2,3} (lanes 16-31).
    v2f afrag[2][8];
#pragma unroll
    for (int t = 0; t < 2; ++t) {
        const int row = t * 16 + laneM;
#pragma unroll
        for (int ks = 0; ks < 8; ++ks) {
            const int col = 4 * ks + (hi ? 2 : 0);
            afrag[t][ks] = *(const v2f*)(Mbig + row * 32 + col);
        }
    }
    // Carry coefficients for the D rows this lane owns: r = t*16 + v + (hi?8:0)
    float ccA[2][8], ccB[2][8];
#pragma unroll
    for (int t = 0; t < 2; ++t)
#pragma unroll
        for (int v = 0; v < 8; ++v) {
            const int r = t * 16 + v + (hi ? 8 : 0);
            ccA[t][v] = cA[r];
            ccB[t][v] = cB[r];
        }

    // Initial state (both lanes of a batch pair hold it redundantly).
    float xp = x0[myBatch * 2 + 0];
    float vp = x0[myBatch * 2 + 1];

    const float* __restrict__ ub = u   + (size_t)myBatch * T * 2;
    float* __restrict__       ob = out + (size_t)myBatch * T * 2;

    // B fragments: B[k][n] = u[batch n][t0 + (k>>1)][k&1] -> flat ub[t0*2 + k]
    auto loadB = [&](v2f (&bf)[8], int t0) {
#pragma unroll
        for (int ks = 0; ks < 8; ++ks) {
            const int c = 4 * ks + (hi ? 2 : 0);
            bf[ks] = __builtin_nontemporal_load((const v2f*)(ub + t0 * 2 + c));
        }
    };

    // Compute one chunk from registers, store outputs, advance carry state.
    auto compute = [&](const v2f (&bf)[8], int t0) {
        v8f acc0 = {0.f, 0.f, 0.f, 0.f, 0.f, 0.f, 0.f, 0.f};
        v8f acc1 = {0.f, 0.f, 0.f, 0.f, 0.f, 0.f, 0.f, 0.f};
#pragma unroll
        for (int ks = 0; ks < 8; ++ks) {
            acc0 = __builtin_amdgcn_wmma_f32_16x16x4_f32(
                false, afrag[0][ks], false, bf[ks], (short)0, acc0, false, false);
            acc1 = __builtin_amdgcn_wmma_f32_16x16x4_f32(
                false, afrag[1][ks], false, bf[ks], (short)0, acc1, false, false);
        }
        // Carry: y[r] += A^(j+1) * X_prev
#pragma unroll
        for (int v = 0; v < 8; ++v) {
            acc0[v] += ccA[0][v] * xp + ccB[0][v] * vp;
            acc1[v] += ccA[1][v] * xp + ccB[1][v] * vp;
        }
        // Stores: each lane's 8 accumulators are contiguous in memory (32B x2).
        float* p0 = ob + t0 * 2 + (hi ? 8 : 0);
        v4f s00 = {acc0[0], acc0[1], acc0[2], acc0[3]};
        v4f s01 = {acc0[4], acc0[5], acc0[6], acc0[7]};
        __builtin_nontemporal_store(s00, (v4f*)p0);
        __builtin_nontemporal_store(s01, (v4f*)p0 + 1);
        float* p1 = ob + t0 * 2 + 16 + (hi ? 8 : 0);
        v4f s10 = {acc1[0], acc1[1], acc1[2], acc1[3]};
        v4f s11 = {acc1[4], acc1[5], acc1[6], acc1[7]};
        __builtin_nontemporal_store(s10, (v4f*)p1);
        __builtin_nontemporal_store(s11, (v4f*)p1 + 1);
        // New carry = y at last chunk step: rows 30/31 = tile1 v=6/7, hi lanes.
        xp = __shfl(acc1[6], laneM + 16, 32);
        vp = __shfl(acc1[7], laneM + 16, 32);
    };

    // Depth-2 software pipeline over 16-step chunks; the steady-state loop
    // retires 3 chunks per iteration so buffer rotation is pure register
    // naming (no copies). Invariant at loop head: b0=chunk(t0), b1=chunk(t0+16)
    // already requested. Requires T % 16 == 0 and T >= 32.
    v2f b0[8], b1[8], b2[8];
    loadB(b0, 0);
    loadB(b1, 16);
    int t0 = 0;
    for (; t0 + 80 <= T; t0 += 48) {
        __builtin_prefetch(ub + (t0 + 80) * 2 + (hi ? 16 : 0), 0, 0);
        loadB(b2, t0 + 32);
        compute(b0, t0);
        __builtin_prefetch(ub + (t0 + 96) * 2 + (hi ? 16 : 0), 0, 0);
        loadB(b0, t0 + 48);
        compute(b1, t0 + 16);
        __builtin_prefetch(ub + (t0 + 112) * 2 + (hi ? 16 : 0), 0, 0);
        loadB(b1, t0 + 64);
        compute(b2, t0 + 32);
    }
    // Epilogue: 2..4 chunks remain; all branches are wave-uniform.
    const int rem = (T - t0) >> 4;
    if (rem >= 3) loadB(b2, t0 + 32);
    compute(b0, t0);
    if (rem == 4) loadB(b0, t0 + 48);
    compute(b1, t0 + 16);
    if (rem >= 3) compute(b2, t0 + 32);
    if (rem == 4) compute(b0, t0 + 48);
}

extern "C" void kernel_launch(void* const* d_in, const int* in_sizes, int n_in,
                              void* d_out, int out_size, void* d_ws, size_t ws_size,
                              hipStream_t stream) {
    const float* u          = (const float*)d_in[0];
    const float* x0         = (const float*)d_in[1];
    const float* k_w        = (const float*)d_in[2];
    const float* b_w        = (const float*)d_in[3];
    const float* input_mat  = (const float*)d_in[4];
    const float* input_mask = (const float*)d_in[5];
    float* out = (float*)d_out;
    float* ws  = (float*)d_ws;

    const int B = in_sizes[1] / 2;              // 4096
    const int T = in_sizes[0] / (2 * B);        // 4096

    rnn_spring_setup_kernel<<<1, 1, 0, stream>>>(k_w, b_w, input_mat, input_mask, ws);

    const int tiles = B / 16;                   // 256 waves, one per workgroup
    rnn_spring_scan_kernel<<<tiles, 32, 0, stream>>>(u, x0, out, ws, T);
}